// MyGRU_24790551232659
// MI455X (gfx1250) — compile-verified
//
#include <hip/hip_runtime.h>
#include <math.h>
#include <stdint.h>

#define DIM_IN 256
#define DIM_H  64
#define BATCH  256
#define SEQ    512
#define AS     260   // padded LDS stride for 16x256 input tile (260%64=4 -> conflict-free A reads)
#define HS     68    // padded LDS stride for 16x64 h / h*r tiles

typedef __attribute__((ext_vector_type(2))) float v2f;
typedef __attribute__((ext_vector_type(8))) float v8f;

static __device__ __forceinline__ v8f wmma_f32(v2f a, v2f b, v8f c) {
    // D = A(16x4 f32) * B(4x16 f32) + C(16x16 f32); emits v_wmma_f32_16x16x4_f32
    return __builtin_amdgcn_wmma_f32_16x16x4_f32(false, a, false, b, (short)0, c, false, false);
}

// ---------------------------------------------------------------------------
// Phase 1: input projections. One workgroup = 16 batches at one fixed t
// (tile aligned with phase-2 consumption). 12 waves = gate(3) x ntile(4).
// xu/xr stored tile-major (exact phase-2 WMMA C-register layout -> b128 reads);
// xi stored to d_out in reference layout (consumed in place by phase 2).
// Input tile staged via CDNA5 async global->LDS (ASYNCcnt).
// ---------------------------------------------------------------------------
__global__ __launch_bounds__(384) void gru_xproj(
    const float* __restrict__ inputs,
    const float* __restrict__ W_input, const float* __restrict__ b_input,
    const float* __restrict__ W_update, const float* __restrict__ b_update,
    const float* __restrict__ W_reset, const float* __restrict__ b_reset,
    float* __restrict__ xu, float* __restrict__ xr, float* __restrict__ xi)
{
    __shared__ float atile[16 * AS];

    const int bt = blockIdx.x >> 9;              // batch tile (0..15)
    const int t  = blockIdx.x & (SEQ - 1);       // timestep (0..511)
    const int bbase = bt * 16;

    // async global->LDS: 16 rows (batches) x 256 f32, 1024 x b128 transfers
    for (int i = threadIdx.x; i < 16 * 64; i += blockDim.x) {
        const int r = i >> 6, c4 = i & 63;
        const float* src = inputs + ((size_t)(bbase + r) * SEQ + t) * DIM_IN + c4 * 4;
        const unsigned ldsoff = (unsigned)(uintptr_t)(&atile[r * AS + c4 * 4]);
        asm volatile("global_load_async_to_lds_b128 %0, %1, off"
                     :: "v"(ldsoff), "v"(src) : "memory");
    }
    asm volatile("s_wait_asynccnt 0" ::: "memory");
    __syncthreads();

    const int wave = threadIdx.x >> 5;
    const int lane = threadIdx.x & 31;
    const int gate = wave >> 2;                  // 0=update 1=reset 2=input
    const int ntile = wave & 3;
    const int m = lane & 15;
    const int hi = lane >> 4;                    // 0: K=k..k+1, 1: K=k+2..k+3
    const int col = ntile * 16 + m;

    const float* Wg = (gate == 0) ? W_update : (gate == 1) ? W_reset : W_input;
    const float* bg = (gate == 0) ? b_update : (gate == 1) ? b_reset : b_input;

    v8f acc = {};
    #pragma unroll 4
    for (int k = 0; k < DIM_IN; k += 4) {
        const int kk = k + hi * 2;
        v2f a, b;
        a.x = atile[m * AS + kk];
        a.y = atile[m * AS + kk + 1];
        b.x = Wg[(size_t)kk * DIM_H + col];
        b.y = Wg[(size_t)(kk + 1) * DIM_H + col];
        acc = wmma_f32(a, b, acc);
    }

    const float bias = bg[col];
    if (gate == 2) {
        // xi -> d_out, reference layout (b, t, h)
        #pragma unroll
        for (int v = 0; v < 8; ++v) {
            const int b = bbase + v + hi * 8;
            xi[((size_t)b * SEQ + t) * DIM_H + col] = acc[v] + bias;
        }
    } else {
        // xu/xr -> scratch, tile-major = phase-2 C-register layout, b128 stores
        float* outg = (gate == 0) ? xu : xr;
        float* dst = outg + (((size_t)bt * SEQ + t) * 4 + ntile) * 256 + lane * 8;
        float4 lo = make_float4(acc[0] + bias, acc[1] + bias, acc[2] + bias, acc[3] + bias);
        float4 hi4 = make_float4(acc[4] + bias, acc[5] + bias, acc[6] + bias, acc[7] + bias);
        ((float4*)dst)[0] = lo;
        ((float4*)dst)[1] = hi4;
    }
}

// ---------------------------------------------------------------------------
// Phase 2: sequential scan. One workgroup per 16-batch tile, 4 waves, each wave
// owns 16 hidden columns. Recurrent weights + h + h*r in LDS. Step t+1 operands
// are prefetched during step t to hide global latency behind WMMA + barriers.
// ---------------------------------------------------------------------------
__global__ __launch_bounds__(128) void gru_scan(
    const float* __restrict__ W_input,
    const float* __restrict__ W_update,
    const float* __restrict__ W_reset,
    const float* __restrict__ xu,
    const float* __restrict__ xr,
    float* __restrict__ out)
{
    __shared__ float Wu[DIM_H * DIM_H];
    __shared__ float Wr[DIM_H * DIM_H];
    __shared__ float Wi[DIM_H * DIM_H];
    __shared__ float hbuf[16 * HS];
    __shared__ float hrbuf[16 * HS];

    for (int i = threadIdx.x; i < DIM_H * DIM_H; i += blockDim.x) {
        Wu[i] = W_update[DIM_IN * DIM_H + i];
        Wr[i] = W_reset [DIM_IN * DIM_H + i];
        Wi[i] = W_input [DIM_IN * DIM_H + i];
    }
    for (int i = threadIdx.x; i < 16 * HS; i += blockDim.x) hbuf[i] = 0.0f;
    __syncthreads();

    const int wave = threadIdx.x >> 5;           // hidden columns [wave*16, wave*16+16)
    const int lane = threadIdx.x & 31;
    const int m = lane & 15;
    const int hi = lane >> 4;
    const int ncol = wave * 16 + m;
    const int bt = blockIdx.x;
    const int bbase = bt * 16;

    // tile-major base for this wave/lane; stride per t = 4*256 floats
    const float* pu_base = xu + ((size_t)bt * SEQ * 4 + wave) * 256 + lane * 8;
    const float* pr_base = xr + ((size_t)bt * SEQ * 4 + wave) * 256 + lane * 8;

    // preload step 0 operands
    float4 cu0, cu1, cr0, cr1;
    float ci[8];
    {
        const float* pu = pu_base;
        const float* pr = pr_base;
        cu0 = ((const float4*)pu)[0]; cu1 = ((const float4*)pu)[1];
        cr0 = ((const float4*)pr)[0]; cr1 = ((const float4*)pr)[1];
        #pragma unroll
        for (int v = 0; v < 8; ++v) {
            const int row = v + hi * 8;
            ci[v] = out[((size_t)(bbase + row) * SEQ) * DIM_H + ncol];
        }
    }

    for (int t = 0; t < SEQ; ++t) {
        // ---- prefetch step t+1 operands (consumed next iteration) ----
        float4 nu0 = {}, nu1 = {}, nr0 = {}, nr1 = {};
        float ni[8] = {};
        if (t + 1 < SEQ) {
            const float* pu = pu_base + (size_t)(t + 1) * 4 * 256;
            const float* pr = pr_base + (size_t)(t + 1) * 4 * 256;
            nu0 = ((const float4*)pu)[0]; nu1 = ((const float4*)pu)[1];
            nr0 = ((const float4*)pr)[0]; nr1 = ((const float4*)pr)[1];
            #pragma unroll
            for (int v = 0; v < 8; ++v) {
                const int row = v + hi * 8;
                ni[v] = out[((size_t)(bbase + row) * SEQ + (t + 1)) * DIM_H + ncol];
            }
        }

        // ---- update & reset gates: h_prev @ W{u,r}_h ----
        v8f au = {}, ar = {};
        #pragma unroll
        for (int k = 0; k < DIM_H; k += 4) {
            const int kk = k + hi * 2;
            v2f a, bu, br;
            a.x  = hbuf[m * HS + kk];
            a.y  = hbuf[m * HS + kk + 1];
            bu.x = Wu[kk * DIM_H + ncol];
            bu.y = Wu[(kk + 1) * DIM_H + ncol];
            br.x = Wr[kk * DIM_H + ncol];
            br.y = Wr[(kk + 1) * DIM_H + ncol];
            au = wmma_f32(a, bu, au);
            ar = wmma_f32(a, br, ar);
        }

        const float xuv[8] = {cu0.x, cu0.y, cu0.z, cu0.w, cu1.x, cu1.y, cu1.z, cu1.w};
        const float xrv[8] = {cr0.x, cr0.y, cr0.z, cr0.w, cr1.x, cr1.y, cr1.z, cr1.w};

        float zv[8], hv[8];
        #pragma unroll
        for (int v = 0; v < 8; ++v) {
            const int row = v + hi * 8;                        // batch row in tile
            const float z = 1.0f / (1.0f + __expf(-(xuv[v] + au[v])));
            const float r = 1.0f / (1.0f + __expf(-(xrv[v] + ar[v])));
            const float h = hbuf[row * HS + ncol];
            zv[v] = z; hv[v] = h;
            hrbuf[row * HS + ncol] = h * r;
        }
        __syncthreads();   // hr complete; all reads of hbuf for this step done

        // ---- candidate: (h*r) @ Wi_h ----
        v8f ai = {};
        #pragma unroll
        for (int k = 0; k < DIM_H; k += 4) {
            const int kk = k + hi * 2;
            v2f a, bi;
            a.x  = hrbuf[m * HS + kk];
            a.y  = hrbuf[m * HS + kk + 1];
            bi.x = Wi[kk * DIM_H + ncol];
            bi.y = Wi[(kk + 1) * DIM_H + ncol];
            ai = wmma_f32(a, bi, ai);
        }

        #pragma unroll
        for (int v = 0; v < 8; ++v) {
            const int row = v + hi * 8;
            const float th = tanhf(ci[v] + ai[v]);
            const float hn = (1.0f - zv[v]) * hv[v] + zv[v] * th;
            out[((size_t)(bbase + row) * SEQ + t) * DIM_H + ncol] = hn;
            hbuf[row * HS + ncol] = hn;
        }
        __syncthreads();   // h_next visible before next step's matmuls

        cu0 = nu0; cu1 = nu1; cr0 = nr0; cr1 = nr1;
        #pragma unroll
        for (int v = 0; v < 8; ++v) ci[v] = ni[v];
    }
}

extern "C" void kernel_launch(void* const* d_in, const int* in_sizes, int n_in,
                              void* d_out, int out_size, void* d_ws, size_t ws_size,
                              hipStream_t stream) {
    const float* inputs   = (const float*)d_in[0];
    const float* W_input  = (const float*)d_in[1];
    const float* b_input  = (const float*)d_in[2];
    const float* W_update = (const float*)d_in[3];
    const float* b_update = (const float*)d_in[4];
    const float* W_reset  = (const float*)d_in[5];
    const float* b_reset  = (const float*)d_in[6];
    float* out = (float*)d_out;

    const size_t G = (size_t)BATCH * SEQ * DIM_H;   // 8.39M floats per gate
    float* xu = (float*)d_ws;                       // needs 2*G*4 = 67 MB scratch
    float* xr = xu + G;

    // Phase 1: input projections (xi -> d_out, consumed in-place by phase 2)
    gru_xproj<<<(BATCH * SEQ) / 16, 384, 0, stream>>>(
        inputs, W_input, b_input, W_update, b_update, W_reset, b_reset,
        xu, xr, out);

    // Phase 2: 512-step recurrence
    gru_scan<<<BATCH / 16, 128, 0, stream>>>(W_input, W_update, W_reset, xu, xr, out);
}